// MemoryAugmentedNetwork_30683246363134
// MI455X (gfx1250) — compile-verified
//
#include <hip/hip_runtime.h>
#include <hip/hip_bf16.h>

// MemoryAugmentedNetwork, MI455X (gfx1250).
// Only the last token's hidden state is live -> skip the 4096-row GEMMs.
// Whole problem is HBM-bound (~323 MB); keys scan (256 MB) done with
// V_WMMA_F32_16X16X4_F32 (exact fp32 math) + nontemporal loads.

#define S_TOK   4096
#define IN_DIM  2048
#define H_DIM   2048
#define D_DIM   1024
#define M_KEYS  65536
#define OUT_DIM 2048
#define EPSF    1e-12f
#define NEG_INF (-3.402823466e38f)

typedef float v2f __attribute__((ext_vector_type(2)));
typedef float v8f __attribute__((ext_vector_type(8)));

// ---------------------------------------------------------------------------
// Generic GEMV: y[j] = (relu?) b[j] + sum_i v[i] * W[i*N + j]   (W row-major KxN)
// v cached in LDS; W streamed once, fully coalesced across the block.
// ---------------------------------------------------------------------------
__global__ __launch_bounds__(256) void gemv_kernel(
    const float* __restrict__ v, const float* __restrict__ W,
    const float* __restrict__ b, float* __restrict__ y,
    int K, int N, int do_relu) {
  __shared__ float sv[3072];
  for (int i = threadIdx.x; i < K; i += blockDim.x) sv[i] = v[i];
  __syncthreads();
  int j = blockIdx.x * blockDim.x + threadIdx.x;
  if (j >= N) return;
  float acc = b[j];
  const float* wp = W + j;
#pragma unroll 8
  for (int i = 0; i < K; ++i) {
    acc = fmaf(sv[i], wp[(size_t)i * N], acc);
  }
  if (do_relu) acc = fmaxf(acc, 0.0f);
  y[j] = acc;
}

// ---------------------------------------------------------------------------
// invq = 1 / max(||q||, eps)
// ---------------------------------------------------------------------------
__global__ __launch_bounds__(256) void qnorm_kernel(const float* __restrict__ q,
                                                    float* __restrict__ invq) {
  __shared__ float red[256];
  float s = 0.0f;
  for (int i = threadIdx.x; i < D_DIM; i += 256) { float t = q[i]; s = fmaf(t, t, s); }
  red[threadIdx.x] = s;
  __syncthreads();
  for (int off = 128; off > 0; off >>= 1) {
    if (threadIdx.x < off) red[threadIdx.x] += red[threadIdx.x + off];
    __syncthreads();
  }
  if (threadIdx.x == 0) invq[0] = 1.0f / fmaxf(sqrtf(red[0]), EPSF);
}

// ---------------------------------------------------------------------------
// Similarity scan over keys with WMMA f32 16x16x4.
// Each wave owns one tile of 16 key rows. Per k-step of 4:
//   A (16x4): row M=0 = q[k..k+3] (lane0 -> K0,K1 ; lane16 -> K2,K3), rest 0
//   B (4x16): B[kk][n] = keys[m0+n][k+kk]; lane n<16 loads keys row n float2 at k,
//             lane n+16 loads the float2 at k+2  (contiguous streaming reads)
//   D row M=0, lane n (VGPR0) accumulates dot(q, key[m0+n]).
// Row sum-of-squares accumulated on the same data; halves merged via shfl_xor(16).
// Block writes its local top-3 of weighted = dot*invq/max(||k||,eps)*importance.
// ---------------------------------------------------------------------------
__global__ __launch_bounds__(256) void sims_topk_kernel(
    const float* __restrict__ q, const float* __restrict__ invq_p,
    const float* __restrict__ keys, const float* __restrict__ importance,
    float* __restrict__ candv, int* __restrict__ candi) {
  __shared__ float sq[D_DIM];
  __shared__ float swv[128];
  __shared__ int   swi[128];
  for (int i = threadIdx.x; i < D_DIM; i += 256) sq[i] = q[i];
  __syncthreads();

  const float invq = invq_p[0];
  const int wave = threadIdx.x >> 5;   // 0..7
  const int lane = threadIdx.x & 31;
  const int row  = lane & 15;
  const int hi   = lane >> 4;          // 0: K{0,1}, 1: K{2,3}
  const long tile = (long)blockIdx.x * 8 + wave;   // 0..4095, exact cover
  const long m0   = tile * 16;

  const float* kp = keys + (m0 + row) * (long)D_DIM + 2 * hi;
  v8f c = {0.f, 0.f, 0.f, 0.f, 0.f, 0.f, 0.f, 0.f};
  float nsq = 0.0f;

#pragma unroll 4
  for (int k = 0; k < D_DIM; k += 4) {
    v2f bk = __builtin_nontemporal_load((const v2f*)(kp + k));   // keys, zero reuse
    float a0 = sq[k + 2 * hi];
    float a1 = sq[k + 2 * hi + 1];
    v2f a;
    a.x = (row == 0) ? a0 : 0.0f;
    a.y = (row == 0) ? a1 : 0.0f;
    nsq = fmaf(bk.x, bk.x, nsq);
    nsq = fmaf(bk.y, bk.y, nsq);
    // D = A(16x4,f32) * B(4x16,f32) + C
    c = __builtin_amdgcn_wmma_f32_16x16x4_f32(false, a, false, bk,
                                              (short)0, c, false, false);
  }
  nsq += __shfl_xor(nsq, 16, 32);      // combine K{0,1} and K{2,3} halves per row

  float w = NEG_INF;
  int   mi = -1;
  if (lane < 16) {
    float dot = c[0];                  // D[M=0][N=lane]
    float sim = dot * invq / fmaxf(sqrtf(nsq), EPSF);
    w  = sim * importance[m0 + lane];
    mi = (int)(m0 + lane);
    swv[wave * 16 + lane] = w;
    swi[wave * 16 + lane] = mi;
  }
  __syncthreads();

  if (threadIdx.x == 0) {
    float t0 = NEG_INF, t1 = NEG_INF, t2 = NEG_INF;
    int   i0 = -1, i1 = -1, i2 = -1;
    for (int i = 0; i < 128; ++i) {
      float v = swv[i]; int ix = swi[i];
      if (v > t0)      { t2 = t1; i2 = i1; t1 = t0; i1 = i0; t0 = v; i0 = ix; }
      else if (v > t1) { t2 = t1; i2 = i1; t1 = v;  i1 = ix; }
      else if (v > t2) { t2 = v;  i2 = ix; }
    }
    candv[blockIdx.x * 3 + 0] = t0; candi[blockIdx.x * 3 + 0] = i0;
    candv[blockIdx.x * 3 + 1] = t1; candi[blockIdx.x * 3 + 1] = i1;
    candv[blockIdx.x * 3 + 2] = t2; candi[blockIdx.x * 3 + 2] = i2;
  }
}

// ---------------------------------------------------------------------------
// Global top-3 over block candidates -> softmax -> retrieved = attn @ values[idx]
// ---------------------------------------------------------------------------
__global__ __launch_bounds__(256) void select_retrieve_kernel(
    const float* __restrict__ candv, const int* __restrict__ candi, int ncand,
    const float* __restrict__ values, float* __restrict__ retrieved) {
  __shared__ float attn[3];
  __shared__ int   sidx[3];
  if (threadIdx.x == 0) {
    float t0 = NEG_INF, t1 = NEG_INF, t2 = NEG_INF;
    int   i0 = -1, i1 = -1, i2 = -1;
    for (int i = 0; i < ncand; ++i) {
      float v = candv[i]; int ix = candi[i];
      if (v > t0)      { t2 = t1; i2 = i1; t1 = t0; i1 = i0; t0 = v; i0 = ix; }
      else if (v > t1) { t2 = t1; i2 = i1; t1 = v;  i1 = ix; }
      else if (v > t2) { t2 = v;  i2 = ix; }
    }
    float e0 = 1.0f;                 // exp(t0 - t0)
    float e1 = __expf(t1 - t0);
    float e2 = __expf(t2 - t0);
    float inv = 1.0f / (e0 + e1 + e2);
    attn[0] = e0 * inv; attn[1] = e1 * inv; attn[2] = e2 * inv;
    sidx[0] = i0; sidx[1] = i1; sidx[2] = i2;
  }
  __syncthreads();
  const float a0 = attn[0], a1 = attn[1], a2 = attn[2];
  const long  i0 = sidx[0], i1 = sidx[1], i2 = sidx[2];
  for (int d = threadIdx.x; d < D_DIM; d += 256) {
    retrieved[d] = a0 * values[i0 * D_DIM + d]
                 + a1 * values[i1 * D_DIM + d]
                 + a2 * values[i2 * D_DIM + d];
  }
}

// ---------------------------------------------------------------------------
extern "C" void kernel_launch(void* const* d_in, const int* in_sizes, int n_in,
                              void* d_out, int out_size, void* d_ws, size_t ws_size,
                              hipStream_t stream) {
  const float* x          = (const float*)d_in[0];
  const float* W1         = (const float*)d_in[1];
  const float* b1         = (const float*)d_in[2];
  const float* W2         = (const float*)d_in[3];
  const float* b2         = (const float*)d_in[4];
  const float* Wq         = (const float*)d_in[5];
  const float* bq         = (const float*)d_in[6];
  const float* Wout       = (const float*)d_in[7];
  const float* bout       = (const float*)d_in[8];
  const float* keys       = (const float*)d_in[9];
  const float* values     = (const float*)d_in[10];
  const float* importance = (const float*)d_in[11];
  float* out = (float*)d_out;

  // workspace layout (floats)
  float* ws    = (float*)d_ws;
  float* h1    = ws;                 // 2048
  float* comb  = ws + 2048;          // 3072: [h (2048) | retrieved (1024)]
  float* qv    = ws + 5120;          // 1024
  float* invq  = ws + 6144;          // 1
  float* candv = ws + 6160;          // 512*3
  int*   candi = (int*)(ws + 6160 + 512 * 3);

  const float* x_last = x + (size_t)(S_TOK - 1) * IN_DIM;

  // h1 = relu(x_last @ W1 + b1)
  gemv_kernel<<<H_DIM / 256, 256, 0, stream>>>(x_last, W1, b1, h1, IN_DIM, H_DIM, 1);
  // h = h1 @ W2 + b2   (into comb[0:2048])
  gemv_kernel<<<H_DIM / 256, 256, 0, stream>>>(h1, W2, b2, comb, H_DIM, H_DIM, 0);
  // q = h @ Wq + bq
  gemv_kernel<<<D_DIM / 256, 256, 0, stream>>>(comb, Wq, bq, qv, H_DIM, D_DIM, 0);
  // invq
  qnorm_kernel<<<1, 256, 0, stream>>>(qv, invq);
  // sims + per-block top-3 : 512 blocks * 8 waves = 4096 tiles of 16 keys
  sims_topk_kernel<<<512, 256, 0, stream>>>(qv, invq, keys, importance, candv, candi);
  // global top-3, softmax, gather -> comb[2048:3072]
  select_retrieve_kernel<<<1, 256, 0, stream>>>(candv, candi, 512 * 3, values,
                                                comb + H_DIM);
  // out = comb @ Wout + bout
  gemv_kernel<<<OUT_DIM / 256, 256, 0, stream>>>(comb, Wout, bout, out,
                                                 H_DIM + D_DIM, OUT_DIM, 0);
}